// ClosebyValuationFunction_2327872275189
// MI455X (gfx1250) — compile-verified
//
#include <hip/hip_runtime.h>
#include <stdint.h>

// ---------------------------------------------------------------------------
// Closeby valuation: out[i] = (|z1[i,4]-z2[i,4]| < 2.0 && |z1[i,5]-z2[i,5]| <= 0.1)
//                              ? 0.99f : 0.01f
// Memory-bound streaming op. CDNA5 path: TDM (tensor_load_to_lds) gathers the
// (x,y) column pairs (8B per row, 24B pitch) into contiguous LDS. Each of the
// block's 8 waves issues its own 512-row descriptor so 8 DMA transfers are in
// flight concurrently (same-wave tensor ops are ordered; cross-wave are not).
// Compute reads LDS conflict-free and streams the result with NT stores.
// ---------------------------------------------------------------------------

typedef __attribute__((ext_vector_type(4))) unsigned int v4u;
typedef __attribute__((ext_vector_type(4))) int          v4i;
typedef __attribute__((ext_vector_type(8))) int          v8i;

#define ROWS_PER_BLOCK 2048
#define THREADS        256
#define WAVES          (THREADS / 32)
#define CHUNK_ROWS     (2 * ROWS_PER_BLOCK / WAVES)   // 512 rows per wave

// Issue one TDM descriptor: gather `nrows` 8-byte elements (cols 4..5 of a
// (nrows,6) f32 tile) with a 24-byte row pitch into a contiguous LDS region.
__device__ __forceinline__ void tdm_gather_pairs(const float* gsrc_pair0,
                                                 unsigned lds_byte_off,
                                                 unsigned nrows) {
  const uint64_t ga = (uint64_t)(uintptr_t)gsrc_pair0;

  // ---- D# group 0 (128b): count/flags | lds_addr | global_addr | type ----
  v4u g0;
  g0.x = 1u;                                   // count=1 (valid), user mode
  g0.y = lds_byte_off;                         // lds_addr (bytes)
  g0.z = (unsigned)(ga & 0xFFFFFFFFu);         // global_addr[31:0]
  g0.w = (unsigned)((ga >> 32) & 0x1FFFFFFu)   // global_addr[56:32]
       | (2u << 30);                           // type = 2 ("image")

  // ---- D# group 1 (256b) ----
  v8i g1;
  g1[0] = (3 << 16);                           // data_size=3 -> 8B elements
  g1[1] = (1 << 16);                           // tensor_dim0 = 1 (bits 63:48)
  g1[2] = (int)((nrows & 0xFFFFu) << 16);      // tensor_dim1[15:0]  (bits 95:80)
  g1[3] = (int)((nrows >> 16) & 0xFFFFu)       // tensor_dim1[31:16] (bits 111:96)
        | (1 << 16);                           // tile_dim0 = 1      (bits 127:112)
  g1[4] = (int)(nrows & 0xFFFFu);              // tile_dim1 = nrows; tile_dim2 = 0
  g1[5] = 3;                                   // tensor_dim0_stride = 3 elems = 24B
  g1[6] = 0;                                   // stride hi / dim1_stride lo
  g1[7] = 0;                                   // dim1_stride hi

  v4i gz = {0, 0, 0, 0};                       // groups 2/3 unused (2-D tensor)
#if defined(__clang_major__) && (__clang_major__ >= 23)
  v8i gz8 = {0, 0, 0, 0, 0, 0, 0, 0};
  __builtin_amdgcn_tensor_load_to_lds(g0, g1, gz, gz, gz8, 0);
#else
  __builtin_amdgcn_tensor_load_to_lds(g0, g1, gz, gz, 0);
#endif
}

__global__ __launch_bounds__(THREADS)
void closeby_valuation_kernel(const float* __restrict__ z1,
                              const float* __restrict__ z2,
                              float* __restrict__ out, int B) {
  // [0 .. ROWS) = z1 pairs, [ROWS .. 2*ROWS) = z2 pairs  (32 KiB total)
  __shared__ float2 tile[2 * ROWS_PER_BLOCK];

  const int base = blockIdx.x * ROWS_PER_BLOCK;

  // --- DMA phase: each wave issues one 512-row gather and waits on it. ---
  // All operands are wave-uniform; TDM ignores EXEC, so the whole wave simply
  // executes the issue. Waves 0..3 cover z1 quarters, waves 4..7 cover z2.
  {
    const int wave      = (int)(threadIdx.x >> 5);   // 0..7, wave-uniform
    const int arr       = wave >> 2;                 // 0 -> z1, 1 -> z2
    const int chunk     = wave & 3;                  // quarter within array
    const int chunkBase = base + chunk * CHUNK_ROWS;
    const int rem       = B - chunkBase;
    const int nrows     = (rem < CHUNK_ROWS) ? (rem < 0 ? 0 : rem) : CHUNK_ROWS;

    if (nrows > 0) {
      // Low 32 bits of a flat LDS pointer are the workgroup-relative offset
      // (ISA aperture mapping).
      const unsigned lds0 = (unsigned)(uintptr_t)(void*)&tile[0];
      const unsigned ldsW =
          lds0 + (unsigned)(arr * ROWS_PER_BLOCK + chunk * CHUNK_ROWS) *
                     (unsigned)sizeof(float2);
      const float* src = (arr == 0 ? z1 : z2) + (size_t)chunkBase * 6 + 4;
      tdm_gather_pairs(src, ldsW, (unsigned)nrows);
      __builtin_amdgcn_s_wait_tensorcnt(0);          // wait for own chunk only
    }
  }
  __syncthreads();

  // --- Compute phase ---
  // Stride-THREADS access: lane L reads LDS bank 2L -> conflict-free b64 reads;
  // output stores are fully coalesced 128B/wave, streamed with NT hint.
#pragma unroll
  for (int j = 0; j < ROWS_PER_BLOCK / THREADS; ++j) {
    const int r   = (int)threadIdx.x + j * THREADS;
    const int row = base + r;
    if (row < B) {
      const float2 a = tile[r];
      const float2 b = tile[ROWS_PER_BLOCK + r];
      const float dx = fabsf(a.x - b.x);
      const float dy = fabsf(a.y - b.y);
      const float v  = ((dx < 2.0f) && (dy <= 0.1f)) ? 0.99f : 0.01f;
      __builtin_nontemporal_store(v, out + row);
    }
  }
}

extern "C" void kernel_launch(void* const* d_in, const int* in_sizes, int n_in,
                              void* d_out, int out_size, void* d_ws, size_t ws_size,
                              hipStream_t stream) {
  const float* z1 = (const float*)d_in[0];
  const float* z2 = (const float*)d_in[1];
  float* out      = (float*)d_out;

  const int B      = in_sizes[0] / 6;           // (B, 6) row-major f32
  const int blocks = (B + ROWS_PER_BLOCK - 1) / ROWS_PER_BLOCK;

  closeby_valuation_kernel<<<blocks, THREADS, 0, stream>>>(z1, z2, out, B);
}